// WindowAttention_23132693856901
// MI455X (gfx1250) — compile-verified
//
#include <hip/hip_runtime.h>
#include <math.h>

typedef __attribute__((ext_vector_type(2))) float v2f;
typedef __attribute__((ext_vector_type(8))) float v8f;
typedef __attribute__((ext_vector_type(4))) unsigned int v4u;
typedef __attribute__((ext_vector_type(8))) int v8i;
typedef __attribute__((ext_vector_type(4))) int v4i;

#define TDIM  5
#define HDIM  60
#define WDIM  108
#define CDIM  512
#define NTOK  (TDIM*HDIM*WDIM)   /* 32400 */
#define QKVN  (3*CDIM)           /* 1536  */
#define NHEAD 4
#define HD    128
#define NWIN  144
#define WTOK  225
#define NKEY  825
#define ATTN_SCALE 0.08838834764831843f  /* 128^-0.5 */

// low 32 bits of a flat pointer into LDS == wave-relative LDS byte offset
#define LDS_OFF(p) ((unsigned)(uintptr_t)(p))

// async global->LDS 16B copy (per-lane addresses), ASYNCcnt-tracked
#define ASYNC_LD16(lds, goff, base, imm)                                     \
  asm volatile("global_load_async_to_lds_b128 %0, %1, %2 offset:" #imm       \
               :: "v"(lds), "v"(goff), "s"(base) : "memory")

#define WAIT_ASYNC0() asm volatile("s_wait_asynccnt 0x0" ::: "memory")

// ---------------------------------------------------------------------------
// TDM: DMA a 2D fp32 tile [rows_tile x 16] (row pitch = 512 floats in memory)
// into LDS with pitch 18 floats (pad 2 dwords every 16 dwords). rows beyond
// rows_tensor read as zero (hardware OOB fill).
// ---------------------------------------------------------------------------
__device__ __forceinline__ void tdm_load_tile(
    unsigned lds_off, const float* gptr, int rows_tensor, int rows_tile)
{
  unsigned long long ga = (unsigned long long)(uintptr_t)gptr;
  v4u g0;
  g0.x = 1u;                                   // count = 1 (valid descriptor)
  g0.y = lds_off;                              // lds_addr (bytes)
  g0.z = (unsigned)(ga & 0xFFFFFFFFu);         // global_addr[31:0]
  g0.w = (unsigned)(ga >> 32) | 0x80000000u;   // global_addr[56:32] | type=2
  v8i g1;
  g1[0] = (2 << 16)      // data_size = 4B
        | (1 << 20)      // pad_enable
        | (3 << 22)      // pad_interval code 3 = every 16 dwords (one row)
        | (1 << 25);     // pad_amount   code 1 = 2 dwords  -> pitch 18 floats
  g1[1] = (int)(16u << 16);                          // tensor_dim0 = 16 (lo16)
  g1[2] = (int)(((unsigned)rows_tensor & 0xFFFFu) << 16);  // tensor_dim1 lo16
  g1[3] = (int)(((unsigned)rows_tensor >> 16) | (16u << 16)); // dim1 hi | tile_dim0=16
  g1[4] = rows_tile;                                 // tile_dim1 | tile_dim2=0
  g1[5] = CDIM;                                      // tensor_dim0_stride = 512
  g1[6] = 0;
  g1[7] = 0;
  v4i z4 = {0, 0, 0, 0};
  v8i z8 = {0, 0, 0, 0, 0, 0, 0, 0};
  // 6-arg variant on this toolchain: (g0, g1, g2, g3, extra, cpol)
  __builtin_amdgcn_tensor_load_to_lds(g0, g1, z4, z4, z8, 0);
}

// ---------------------------------------------------------------------------
// Fused GEMM + bias:  C[M,N] = A[M,512] @ Wt[N,512]^T + bias[N]   (fp32)
// Block 256 thr = 8 waves (2x4), wave computes 32x32 via 2x2 f32-WMMA accums.
// Tiles staged by TDM (double-buffered); wave0 issues, TENSORcnt-paced.
// ---------------------------------------------------------------------------
__global__ __launch_bounds__(256) void gemm_bias_kernel(
    const float* __restrict__ A, const float* __restrict__ Wt,
    const float* __restrict__ bias, float* __restrict__ C,
    int M, int N)
{
  __shared__ __align__(16) float As[2][64][18];    // 64 x 16 tile, pitch 18
  __shared__ __align__(16) float Bs[2][128][18];   // 128(n) x 16(k), pitch 18

  const int m0   = blockIdx.x * 64;
  const int n0   = blockIdx.y * 128;
  const int tid  = threadIdx.x;
  const int wave = tid >> 5;
  const int lane = tid & 31;
  const int lh   = lane & 15;
  const int hi   = lane >> 4;      // 0 = lanes 0-15, 1 = lanes 16-31
  const int wm   = wave & 1;       // row block (32 rows)
  const int wn   = wave >> 1;      // col block (32 cols)

  int mrows = M - m0; if (mrows > 64) mrows = 64;
  const bool issuer = (tid < 32);  // wave 0 drives the TDM

  v8f acc[2][2];
  #pragma unroll
  for (int i = 0; i < 2; ++i)
    #pragma unroll
    for (int j = 0; j < 2; ++j)
      #pragma unroll
      for (int e = 0; e < 8; ++e) acc[i][j][e] = 0.0f;

  if (issuer) {
    tdm_load_tile(LDS_OFF(&As[0][0][0]), A + (size_t)m0 * CDIM, mrows, 64);
    tdm_load_tile(LDS_OFF(&Bs[0][0][0]), Wt + (size_t)n0 * CDIM, 128, 128);
  }

  const int NIT = CDIM / 16;   // 32 k-tiles
  #pragma unroll 1
  for (int it = 0; it < NIT; ++it) {
    const int cur = it & 1;
    if (issuer) {
      if (it + 1 < NIT) {
        const int k0n = (it + 1) * 16;
        tdm_load_tile(LDS_OFF(&As[cur ^ 1][0][0]),
                      A + (size_t)m0 * CDIM + k0n, mrows, 64);
        tdm_load_tile(LDS_OFF(&Bs[cur ^ 1][0][0]),
                      Wt + (size_t)n0 * CDIM + k0n, 128, 128);
        __builtin_amdgcn_s_wait_tensorcnt(2);  // current tile done (in-order)
      } else {
        __builtin_amdgcn_s_wait_tensorcnt(0);
      }
    }
    __syncthreads();            // tile visible to all waves

    #pragma unroll
    for (int k4 = 0; k4 < 16; k4 += 4) {
      v2f a[2], b[2];
      #pragma unroll
      for (int im = 0; im < 2; ++im)
        a[im] = *(const v2f*)&As[cur][wm * 32 + im * 16 + lh][k4 + hi * 2];
      #pragma unroll
      for (int in = 0; in < 2; ++in)
        b[in] = *(const v2f*)&Bs[cur][wn * 32 + in * 16 + lh][k4 + hi * 2];
      #pragma unroll
      for (int im = 0; im < 2; ++im)
        #pragma unroll
        for (int in = 0; in < 2; ++in)
          acc[im][in] = __builtin_amdgcn_wmma_f32_16x16x4_f32(
              false, a[im], false, b[in], (short)0, acc[im][in], false, false);
    }
    __syncthreads();            // done reading before next TDM overwrite
  }

  #pragma unroll
  for (int im = 0; im < 2; ++im)
    #pragma unroll
    for (int in = 0; in < 2; ++in) {
      int colg = n0 + wn * 32 + in * 16 + lh;
      float bv = bias[colg];
      #pragma unroll
      for (int i = 0; i < 8; ++i) {
        int rowg = m0 + wm * 32 + im * 16 + i + hi * 8;
        if (rowg < M) C[(size_t)rowg * N + colg] = acc[im][in][i] + bv;
      }
    }
}

// ---------------------------------------------------------------------------
// Rolled-key index decode: kk in [0,825) -> source token id for window (hw,ww)
// ---------------------------------------------------------------------------
__device__ __forceinline__ int key_token(int hw, int ww, int kk)
{
  int t, r, c, dr = 0, dc = 0;
  if (kk < WTOK) {
    t = kk / 45; int p = kk % 45; r = p / 9; c = p % 9;
  } else {
    int j = kk - WTOK; t = j / 120; int ji = j % 120;
    int s = ji / 30, q = ji % 30;
    int sr = s >> 1, sc = s & 1;
    if (!sr) {           // zero block rows 0..2
      if (q < 12) { r = q / 4;      c = (sc ? 0 : 5) + (q & 3); }
      else        { q -= 12; r = 3 + q / 9; c = q % 9; }
    } else {             // zero block rows 2..4
      if (q < 18) { r = q / 9;      c = q % 9; }
      else        { q -= 18; r = 2 + q / 4; c = (sc ? 0 : 5) + (q & 3); }
    }
    dr = sr ? 2 : -2;    // roll shifts: (-2,-4),(-2,4),(2,-4),(2,4)
    dc = sc ? 4 : -4;
  }
  int h = hw * 5 + r - dr; h += HDIM; h %= HDIM;
  int w = ww * 9 + c - dc; w += WDIM; w %= WDIM;
  return t * (HDIM * WDIM) + h * WDIM + w;
}

__device__ __forceinline__ float xormax16(float v) {
  v = fmaxf(v, __shfl_xor(v, 1));
  v = fmaxf(v, __shfl_xor(v, 2));
  v = fmaxf(v, __shfl_xor(v, 4));
  v = fmaxf(v, __shfl_xor(v, 8));
  return v;
}
__device__ __forceinline__ float xorsum16(float v) {
  v += __shfl_xor(v, 1); v += __shfl_xor(v, 2);
  v += __shfl_xor(v, 4); v += __shfl_xor(v, 8);
  return v;
}

// ---------------------------------------------------------------------------
// Streaming (flash-style) attention. Block = 128 thr (4 waves).
// grid = (qblock 0..3, head 0..3, window 0..143). Wave owns 16 queries.
// K/V key tiles gathered with async global->LDS b128 copies (ASYNCcnt).
// ---------------------------------------------------------------------------
__global__ __launch_bounds__(128) void attn_kernel(
    const float* __restrict__ qkv, float* __restrict__ attn)
{
  __shared__ __align__(16) float Kt[16][132];
  __shared__ __align__(16) float Vt[16][132];
  __shared__ __align__(16) float Pw[4][16][18];

  const int qb   = blockIdx.x;
  const int head = blockIdx.y;
  const int win  = blockIdx.z;
  const int tid  = threadIdx.x;
  const int wave = tid >> 5, lane = tid & 31;
  const int lh   = lane & 15, hi = lane >> 4;
  const int hw   = win / 12, ww = win % 12;
  const unsigned long long qkv_base = (unsigned long long)(uintptr_t)qkv;

  // ---- preload Q A-fragments (pre-scaled by 1/sqrt(HD)) ----
  int qrow = qb * 64 + wave * 16 + lh;
  int qr = (qrow < WTOK) ? qrow : (WTOK - 1);
  int qt = qr / 45, qp = qr % 45;
  int qtok = qt * (HDIM * WDIM) + (hw * 5 + qp / 9) * WDIM + (ww * 9 + qp % 9);
  const float* qbase = qkv + (size_t)qtok * QKVN + head * HD;
  v2f qa[32];
  #pragma unroll
  for (int kk = 0; kk < 32; ++kk) {
    v2f t = *(const v2f*)(qbase + kk * 4 + hi * 2);
    t.x *= ATTN_SCALE; t.y *= ATTN_SCALE;
    qa[kk] = t;
  }

  v8f O[8];
  #pragma unroll
  for (int nt = 0; nt < 8; ++nt)
    #pragma unroll
    for (int e = 0; e < 8; ++e) O[nt][e] = 0.0f;
  float run_m[8], run_l[8];
  #pragma unroll
  for (int i = 0; i < 8; ++i) { run_m[i] = -__builtin_inff(); run_l[i] = 0.0f; }

  // per-thread gather slot: key j (16 keys/tile), dim group g (16 floats)
  const int j = tid >> 3;
  const int g = tid & 7;
  const unsigned kd = LDS_OFF(&Kt[j][g * 16]);
  const unsigned vd = LDS_OFF(&Vt[j][g * 16]);

  const int nkt = (NKEY + 15) / 16;   // 52 key tiles
  #pragma unroll 1
  for (int kt = 0; kt < nkt; ++kt) {
    { // ---- async-gather K/V tile (16 keys x 128 dims) into LDS ----
      int kk = kt * 16 + j; if (kk >= NKEY) kk = NKEY - 1;
      int tok = key_token(hw, ww, kk);
      unsigned koff = (unsigned)tok * (QKVN * 4u)
                    + (CDIM + head * HD + g * 16) * 4u;   // byte offset of K
      unsigned voff = koff + CDIM * 4u;                   // V is next 512 cols
      ASYNC_LD16(kd, koff, qkv_base, 0);
      ASYNC_LD16(kd, koff, qkv_base, 16);
      ASYNC_LD16(kd, koff, qkv_base, 32);
      ASYNC_LD16(kd, koff, qkv_base, 48);
      ASYNC_LD16(vd, voff, qkv_base, 0);
      ASYNC_LD16(vd, voff, qkv_base, 16);
      ASYNC_LD16(vd, voff, qkv_base, 32);
      ASYNC_LD16(vd, voff, qkv_base, 48);
      WAIT_ASYNC0();
    }
    __syncthreads();

    // ---- S = (Q*scale) . K^T  (16q x 16k), K=128 in 32 WMMA steps ----
    v8f s;
    #pragma unroll
    for (int e = 0; e < 8; ++e) s[e] = 0.0f;
    #pragma unroll
    for (int kk = 0; kk < 32; ++kk) {
      v2f b = *(const v2f*)&Kt[lh][kk * 4 + hi * 2];
      s = __builtin_amdgcn_wmma_f32_16x16x4_f32(
          false, qa[kk], false, b, (short)0, s, false, false);
    }

    // ---- online softmax (rows i / i+8 split across lane halves) ----
    const bool kvalid = (kt * 16 + lh) < NKEY;
    #pragma unroll
    for (int i = 0; i < 8; ++i) {
      float sv   = kvalid ? s[i] : -__builtin_inff();
      float mx   = xormax16(sv);
      float mnew = fmaxf(run_m[i], mx);
      float p    = __expf(sv - mnew);
      float sum  = xorsum16(p);
      float corr = __expf(run_m[i] - mnew);
      run_l[i] = run_l[i] * corr + sum;
      run_m[i] = mnew;
      #pragma unroll
      for (int nt = 0; nt < 8; ++nt) O[nt][i] *= corr;
      Pw[wave][i + hi * 8][lh] = p;   // C-layout -> LDS (same-wave DS order)
    }

    // ---- O += P (16x16) x V (16x128) : 4 k-steps x 8 n-tiles ----
    #pragma unroll
    for (int kk = 0; kk < 4; ++kk) {
      v2f pa = *(const v2f*)&Pw[wave][lh][kk * 4 + hi * 2];
      int krow = kk * 4 + hi * 2;
      #pragma unroll
      for (int nt = 0; nt < 8; ++nt) {
        int cc = nt * 16 + lh;
        v2f b; b.x = Vt[krow][cc]; b.y = Vt[krow + 1][cc];
        O[nt] = __builtin_amdgcn_wmma_f32_16x16x4_f32(
            false, pa, false, b, (short)0, O[nt], false, false);
      }
    }
    __syncthreads();   // all waves done with Kt/Vt before next gather
  }

  // ---- epilogue: normalize and store (window-major token order) ----
  const int q0w = qb * 64 + wave * 16;
  #pragma unroll
  for (int i = 0; i < 8; ++i) {
    int row = q0w + i + hi * 8;
    if (row < WTOK) {
      float inv = 1.0f / run_l[i];
      #pragma unroll
      for (int nt = 0; nt < 8; ++nt) {
        int col = head * HD + nt * 16 + lh;
        attn[(size_t)(win * WTOK + row) * CDIM + col] = O[nt][i] * inv;
      }
    }
  }
}

// ---------------------------------------------------------------------------
extern "C" void kernel_launch(void* const* d_in, const int* in_sizes, int n_in,
                              void* d_out, int out_size, void* d_ws, size_t ws_size,
                              hipStream_t stream)
{
  (void)in_sizes; (void)n_in; (void)out_size; (void)ws_size;
  const float* x      = (const float*)d_in[0];
  const float* qkv_w  = (const float*)d_in[1];
  const float* qkv_b  = (const float*)d_in[2];
  const float* proj_w = (const float*)d_in[3];
  const float* proj_b = (const float*)d_in[4];
  float* out = (float*)d_out;

  float* qkv     = (float*)d_ws;                       // 32400 x 1536 f32
  float* attnbuf = qkv + (size_t)NTOK * QKVN;          // 32400 x 512  f32

  dim3 blk(256);
  dim3 g1((NTOK + 63) / 64, QKVN / 128);
  hipLaunchKernelGGL(gemm_bias_kernel, g1, blk, 0, stream,
                     x, qkv_w, qkv_b, qkv, NTOK, QKVN);

  dim3 g2(4, NHEAD, NWIN);
  hipLaunchKernelGGL(attn_kernel, g2, dim3(128), 0, stream, qkv, attnbuf);

  dim3 g3((NTOK + 63) / 64, CDIM / 128);
  hipLaunchKernelGGL(gemm_bias_kernel, g3, blk, 0, stream,
                     attnbuf, proj_w, proj_b, out, NTOK, CDIM);
}